// GRU_DFA_5394478924445
// MI455X (gfx1250) — compile-verified
//
#include <hip/hip_runtime.h>
#include <math.h>

typedef __attribute__((ext_vector_type(2))) float v2f;
typedef __attribute__((ext_vector_type(8))) float v8f;

#define Bsz 64
#define Tn  256
#define In  128
#define Hn  512
#define On  64
#define BH  (Bsz*Hn)          /* 32768  */
#define TB  (Tn*Bsz)          /* 16384  */
#define TBH (TB*Hn)           /* 8388608 */
#define LDS_STRIDE 516        /* 512 + 4 pad: bank-conflict-free row stride */

// ---------------- WMMA f32 16x16x4 helpers ----------------
__device__ __forceinline__ v8f wmma_f32(v2f a, v2f b, v8f c) {
  return __builtin_amdgcn_wmma_f32_16x16x4_f32(false, a, false, b, (short)0, c,
                                               false, false);
}
// A fragment, A [M,K] row-major. Lanes 0-15: M=lane,{K=k0,k0+1}; 16-31:{k0+2,k0+3}
__device__ __forceinline__ v2f ldA(const float* A, int lda, int m0, int k0, int lane) {
  int r = lane & 15, ko = (lane >> 4) << 1;
  const float* p = A + (size_t)(m0 + r) * lda + (k0 + ko);
  v2f v; v.x = p[0]; v.y = p[1]; return v;
}
// A fragment, logical A[m][k] = M[k*lda + m]  (A-transposed source)
__device__ __forceinline__ v2f ldAT(const float* A, int lda, int m0, int k0, int lane) {
  int r = lane & 15, ko = (lane >> 4) << 1;
  const float* p = A + (size_t)(k0 + ko) * lda + (m0 + r);
  v2f v; v.x = p[0]; v.y = p[lda]; return v;
}
// B fragment, logical B[k][n] = Bm[n*ldb + k]  (Bm stored [N,K]; C = A@Bm^T)
__device__ __forceinline__ v2f ldBT(const float* Bm, int ldb, int n0, int k0, int lane) {
  int c = lane & 15, ko = (lane >> 4) << 1;
  const float* p = Bm + (size_t)(n0 + c) * ldb + (k0 + ko);
  v2f v; v.x = p[0]; v.y = p[1]; return v;
}
// B fragment, Bm stored [K,N] row-major
__device__ __forceinline__ v2f ldB(const float* Bm, int ldb, int n0, int k0, int lane) {
  int c = lane & 15, ko = (lane >> 4) << 1;
  const float* p = Bm + (size_t)(k0 + ko) * ldb + (n0 + c);
  v2f v; v.x = p[0]; v.y = p[ldb]; return v;
}
// B fragment from padded LDS slice (rows local, stride LDS_STRIDE)
__device__ __forceinline__ v2f ldBT_lds(const float* S, int n0l, int k0, int lane) {
  int c = lane & 15, ko = (lane >> 4) << 1;
  const float* p = S + (size_t)(n0l + c) * LDS_STRIDE + (k0 + ko);
  v2f v; v.x = p[0]; v.y = p[1]; return v;
}
__device__ __forceinline__ void store_tile(float* C, int ldc, int m0, int n0, int lane,
                                           const v8f& acc, const float* bias, float alpha) {
  int c = lane & 15, mo = (lane >> 4) * 8;
#pragma unroll
  for (int j = 0; j < 8; ++j) {
    int n = n0 + c;
    float v = alpha * acc[j];
    if (bias) v += bias[n];
    C[(size_t)(m0 + mo + j) * ldc + n] = v;
  }
}

// ---------------- device-wide barrier (for persistent scan) ----------------
__device__ __forceinline__ void grid_sync(int* cnt, int* gen, int nblk) {
  __threadfence();
  __syncthreads();
  if (threadIdx.x == 0) {
    int g = __hip_atomic_load(gen, __ATOMIC_ACQUIRE, __HIP_MEMORY_SCOPE_AGENT);
    int prev = __hip_atomic_fetch_add(cnt, 1, __ATOMIC_ACQ_REL, __HIP_MEMORY_SCOPE_AGENT);
    if (prev == nblk - 1) {
      __hip_atomic_store(cnt, 0, __ATOMIC_RELAXED, __HIP_MEMORY_SCOPE_AGENT);
      __hip_atomic_fetch_add(gen, 1, __ATOMIC_RELEASE, __HIP_MEMORY_SCOPE_AGENT);
    } else {
      while (__hip_atomic_load(gen, __ATOMIC_ACQUIRE, __HIP_MEMORY_SCOPE_AGENT) == g) {
        __builtin_amdgcn_s_sleep(8);
      }
    }
  }
  __syncthreads();
}

// -------- GEMM: C = alpha*(A @ Bm^T) (+bias), 4 N-tiles/wave, SW-pipelined --------
__global__ void gemm_abt4(const float* __restrict__ A, int lda,
                          const float* __restrict__ Bm, int ldb,
                          float* __restrict__ C, int ldc,
                          const float* __restrict__ bias, int K, float alpha) {
  int lane = threadIdx.x & 31, wave = threadIdx.x >> 5;
  int m0 = blockIdx.x * 16;
  int n0 = (blockIdx.y * 4 + wave) * 64;
  v8f acc0 = {}, acc1 = {}, acc2 = {}, acc3 = {};
  v2f aC[2], b0C[2], b1C[2], b2C[2], b3C[2];
#pragma unroll
  for (int u = 0; u < 2; ++u) {
    int k = 4 * u;
    aC[u]  = ldA(A, lda, m0, k, lane);
    b0C[u] = ldBT(Bm, ldb, n0,      k, lane);
    b1C[u] = ldBT(Bm, ldb, n0 + 16, k, lane);
    b2C[u] = ldBT(Bm, ldb, n0 + 32, k, lane);
    b3C[u] = ldBT(Bm, ldb, n0 + 48, k, lane);
  }
  for (int k = 0; k < K; k += 8) {
    v2f aN[2], b0N[2], b1N[2], b2N[2], b3N[2];
    if (k + 8 < K) {
#pragma unroll
      for (int u = 0; u < 2; ++u) {
        int kn = k + 8 + 4 * u;
        aN[u]  = ldA(A, lda, m0, kn, lane);
        b0N[u] = ldBT(Bm, ldb, n0,      kn, lane);
        b1N[u] = ldBT(Bm, ldb, n0 + 16, kn, lane);
        b2N[u] = ldBT(Bm, ldb, n0 + 32, kn, lane);
        b3N[u] = ldBT(Bm, ldb, n0 + 48, kn, lane);
      }
    }
#pragma unroll
    for (int u = 0; u < 2; ++u) {
      acc0 = wmma_f32(aC[u], b0C[u], acc0);
      acc1 = wmma_f32(aC[u], b1C[u], acc1);
      acc2 = wmma_f32(aC[u], b2C[u], acc2);
      acc3 = wmma_f32(aC[u], b3C[u], acc3);
    }
#pragma unroll
    for (int u = 0; u < 2; ++u) {
      aC[u] = aN[u]; b0C[u] = b0N[u]; b1C[u] = b1N[u]; b2C[u] = b2N[u]; b3C[u] = b3N[u];
    }
  }
  store_tile(C, ldc, m0, n0,      lane, acc0, bias, alpha);
  store_tile(C, ldc, m0, n0 + 16, lane, acc1, bias, alpha);
  store_tile(C, ldc, m0, n0 + 32, lane, acc2, bias, alpha);
  store_tile(C, ldc, m0, n0 + 48, lane, acc3, bias, alpha);
}

// -------- GEMM: C = alpha*(A^T @ B), 4 N-tiles/wave, SW-pipelined --------
__global__ void gemm_atb4(const float* __restrict__ A, int lda,
                          const float* __restrict__ Bm, int ldb,
                          float* __restrict__ C, int ldc, int K, float alpha) {
  int lane = threadIdx.x & 31, wave = threadIdx.x >> 5;
  int m0 = blockIdx.x * 16;
  int n0 = (blockIdx.y * 4 + wave) * 64;
  v8f acc0 = {}, acc1 = {}, acc2 = {}, acc3 = {};
  v2f aC[2], b0C[2], b1C[2], b2C[2], b3C[2];
#pragma unroll
  for (int u = 0; u < 2; ++u) {
    int k = 4 * u;
    aC[u]  = ldAT(A, lda, m0, k, lane);
    b0C[u] = ldB(Bm, ldb, n0,      k, lane);
    b1C[u] = ldB(Bm, ldb, n0 + 16, k, lane);
    b2C[u] = ldB(Bm, ldb, n0 + 32, k, lane);
    b3C[u] = ldB(Bm, ldb, n0 + 48, k, lane);
  }
  for (int k = 0; k < K; k += 8) {
    v2f aN[2], b0N[2], b1N[2], b2N[2], b3N[2];
    if (k + 8 < K) {
#pragma unroll
      for (int u = 0; u < 2; ++u) {
        int kn = k + 8 + 4 * u;
        aN[u]  = ldAT(A, lda, m0, kn, lane);
        b0N[u] = ldB(Bm, ldb, n0,      kn, lane);
        b1N[u] = ldB(Bm, ldb, n0 + 16, kn, lane);
        b2N[u] = ldB(Bm, ldb, n0 + 32, kn, lane);
        b3N[u] = ldB(Bm, ldb, n0 + 48, kn, lane);
      }
    }
#pragma unroll
    for (int u = 0; u < 2; ++u) {
      acc0 = wmma_f32(aC[u], b0C[u], acc0);
      acc1 = wmma_f32(aC[u], b1C[u], acc1);
      acc2 = wmma_f32(aC[u], b2C[u], acc2);
      acc3 = wmma_f32(aC[u], b3C[u], acc3);
    }
#pragma unroll
    for (int u = 0; u < 2; ++u) {
      aC[u] = aN[u]; b0C[u] = b0N[u]; b1C[u] = b1N[u]; b2C[u] = b2N[u]; b3C[u] = b3N[u];
    }
  }
  store_tile(C, ldc, m0, n0,      lane, acc0, nullptr, alpha);
  store_tile(C, ldc, m0, n0 + 16, lane, acc1, nullptr, alpha);
  store_tile(C, ldc, m0, n0 + 32, lane, acc2, nullptr, alpha);
  store_tile(C, ldc, m0, n0 + 48, lane, acc3, nullptr, alpha);
}

// -------- single-tile A^T@B for the narrow dV GEMMs (N=128), SW-pipelined --------
__global__ void gemm_atb(const float* __restrict__ A, int lda,
                         const float* __restrict__ Bm, int ldb,
                         float* __restrict__ C, int ldc, int K, float alpha) {
  int lane = threadIdx.x & 31, wave = threadIdx.x >> 5;
  int m0 = blockIdx.x * 16;
  int n0 = (blockIdx.y * 4 + wave) * 16;
  v8f acc = {};
  v2f aC[4], bC[4];
#pragma unroll
  for (int u = 0; u < 4; ++u) {
    aC[u] = ldAT(A, lda, m0, 4 * u, lane);
    bC[u] = ldB(Bm, ldb, n0, 4 * u, lane);
  }
  for (int k = 0; k < K; k += 16) {
    v2f aN[4], bN[4];
    if (k + 16 < K) {
#pragma unroll
      for (int u = 0; u < 4; ++u) {
        aN[u] = ldAT(A, lda, m0, k + 16 + 4 * u, lane);
        bN[u] = ldB(Bm, ldb, n0, k + 16 + 4 * u, lane);
      }
    }
#pragma unroll
    for (int u = 0; u < 4; ++u) acc = wmma_f32(aC[u], bC[u], acc);
#pragma unroll
    for (int u = 0; u < 4; ++u) { aC[u] = aN[u]; bC[u] = bN[u]; }
  }
  store_tile(C, ldc, m0, n0, lane, acc, nullptr, alpha);
}

// ---------------- Persistent GRU scan ----------------
// grid MUST be <<<64, 128, 64*LDS_STRIDE*4>>> : 256 waves.
// Block bx: gate g=bx>>5 (0:z/W2, 1:r/W3), mt=(bx>>3)&3, cols 64*(bx&7)..+63.
// Each wave: one 16x16 tile. Gate W slice staged in LDS once (time-invariant).
// Phase 2 (blocks 0..31, same tile coords): ht=tanh(hr@W1^T+Xv1), hn blend.
__global__ void scan_kernel(const float* __restrict__ W1,
                            const float* __restrict__ W2, const float* __restrict__ W2b,
                            const float* __restrict__ W3, const float* __restrict__ W3b,
                            const float* __restrict__ Xv1, const float* __restrict__ Xv2,
                            const float* __restrict__ Xv3,
                            float* __restrict__ zs, float* __restrict__ rs,
                            float* __restrict__ hs, float* __restrict__ hid,
                            const float* __restrict__ h0, float* __restrict__ hr,
                            int* bar_cnt, int* bar_gen) {
  extern __shared__ float ldsW[];        // [64][LDS_STRIDE]
  const int lane = threadIdx.x & 31;
  const int w = threadIdx.x >> 5;        // wave in block: 0..3
  const int bx = blockIdx.x;             // 0..63
  const int nblk = gridDim.x;
  const int g = bx >> 5;
  const int mt = (bx >> 3) & 3;
  const int nbase = 64 * (bx & 7);
  const int m0 = mt * 16;
  const int n0 = nbase + 16 * w;
  const int n0l = 16 * w;
  const int c = lane & 15, mo = (lane >> 4) * 8;

  // one-time stage of this block's gate-weight slice into padded LDS
  {
    const float* Wsrc = (g ? W3 : W2) + (size_t)nbase * Hn;
    for (int idx = threadIdx.x; idx < 64 * 128; idx += 128) {
      int r = idx >> 7, q = idx & 127;
      ((float4*)(ldsW + (size_t)r * LDS_STRIDE))[q] =
          ((const float4*)(Wsrc + (size_t)r * Hn))[q];
    }
  }
  __syncthreads();

  const float* bias = g ? W3b : W2b;
  const float* XvG  = g ? Xv3 : Xv2;
  float* outG = g ? rs : zs;

  for (int t = 0; t < Tn; ++t) {
    const float* hp = t ? hid + (size_t)(t - 1) * BH : h0;

    // ---- phase 1: gate tile; A (global) double-buffered, B from LDS ----
    {
      v8f acc = {};
      v2f aC[4];
#pragma unroll
      for (int u = 0; u < 4; ++u) aC[u] = ldA(hp, Hn, m0, 4 * u, lane);
      for (int k = 0; k < Hn; k += 16) {
        v2f aN[4];
        if (k + 16 < Hn) {
#pragma unroll
          for (int u = 0; u < 4; ++u) aN[u] = ldA(hp, Hn, m0, k + 16 + 4 * u, lane);
        }
#pragma unroll
        for (int u = 0; u < 4; ++u) {
          v2f b = ldBT_lds(ldsW, n0l, k + 4 * u, lane);
          acc = wmma_f32(aC[u], b, acc);
        }
#pragma unroll
        for (int u = 0; u < 4; ++u) aC[u] = aN[u];
      }
      const float* Xv = XvG + (size_t)t * BH;
      float* outp = outG + (size_t)t * BH;
#pragma unroll
      for (int j = 0; j < 8; ++j) {
        size_t idx = (size_t)(m0 + mo + j) * Hn + (n0 + c);
        float v = acc[j] + bias[n0 + c] + Xv[idx];
        float sg = 1.0f / (1.0f + __expf(-v));
        outp[idx] = sg;
        if (g) hr[idx] = sg * hp[idx];
      }
    }
    grid_sync(bar_cnt, bar_gen, nblk);

    // ---- phase 2: candidate + blend (blocks 0..31); A and B double-buffered ----
    if (bx < 32) {
      v8f acc = {};
      v2f aC[4], bC[4];
#pragma unroll
      for (int u = 0; u < 4; ++u) {
        aC[u] = ldA(hr, Hn, m0, 4 * u, lane);
        bC[u] = ldBT(W1, Hn, n0, 4 * u, lane);
      }
      for (int k = 0; k < Hn; k += 16) {
        v2f aN[4], bN[4];
        if (k + 16 < Hn) {
#pragma unroll
          for (int u = 0; u < 4; ++u) {
            aN[u] = ldA(hr, Hn, m0, k + 16 + 4 * u, lane);
            bN[u] = ldBT(W1, Hn, n0, k + 16 + 4 * u, lane);
          }
        }
#pragma unroll
        for (int u = 0; u < 4; ++u) acc = wmma_f32(aC[u], bC[u], acc);
#pragma unroll
        for (int u = 0; u < 4; ++u) { aC[u] = aN[u]; bC[u] = bN[u]; }
      }
      const float* Xv = Xv1 + (size_t)t * BH;
      const float* zst = zs + (size_t)t * BH;
      float* hst = hs + (size_t)t * BH;
      float* hidt = hid + (size_t)t * BH;
#pragma unroll
      for (int j = 0; j < 8; ++j) {
        size_t idx = (size_t)(m0 + mo + j) * Hn + (n0 + c);
        float ht = tanhf(acc[j] + Xv[idx]);
        float z = zst[idx];
        hst[idx] = ht;
        hidt[idx] = (1.0f - z) * ht + z * hp[idx];
      }
    }
    grid_sync(bar_cnt, bar_gen, nblk);
  }
}

// ---------------- Small / elementwise kernels ----------------
__global__ void zero_kernel(float* p, int n) {
  int i = blockIdx.x * blockDim.x + threadIdx.x;
  if (i < n) p[i] = 0.0f;
}
__global__ void copy_kernel(const float* s, float* d, int n) {
  int i = blockIdx.x * blockDim.x + threadIdx.x;
  if (i < n) d[i] = s[i];
}
__global__ void xmat_kernel(const float* __restrict__ x, float* __restrict__ xmat) {
  int i = blockIdx.x * blockDim.x + threadIdx.x;
  if (i >= TB * In) return;
  int icol = i & (In - 1);
  int m = i >> 7;
  int b = m & (Bsz - 1);
  int t = m >> 6;
  xmat[i] = x[((size_t)b * Tn + t) * In + icol];
}
// hprev[t]: t==0 -> hid[T-1]; t==1 -> 0; else hid[t-2]   (reference's list[-1] quirk)
__global__ void hprev_kernel(const float* __restrict__ hid, float* __restrict__ hp) {
  int i = blockIdx.x * blockDim.x + threadIdx.x;
  if (i >= TBH) return;
  int t = i >> 15;
  int rb = i & (BH - 1);
  float v;
  if (t == 0)      v = hid[(size_t)(Tn - 1) * BH + rb];
  else if (t == 1) v = 0.0f;
  else             v = hid[(size_t)(t - 2) * BH + rb];
  hp[i] = v;
}
__global__ void head_kernel(const float* __restrict__ hT, const float* __restrict__ Ww,
                            const float* __restrict__ Wb, const int* __restrict__ y,
                            float* __restrict__ outp, float* __restrict__ errp) {
  int b = blockIdx.x, o = threadIdx.x;
  float acc = Wb[o];
  const float* h = hT + (size_t)b * Hn;
  const float* w = Ww + (size_t)o * Hn;
  for (int k = 0; k < Hn; ++k) acc += h[k] * w[k];
  __shared__ float sl[On];
  sl[o] = acc; __syncthreads();
  float mx = sl[0];
  for (int i = 1; i < On; ++i) mx = fmaxf(mx, sl[i]);
  float e = __expf(acc - mx);
  __shared__ float se[On];
  se[o] = e; __syncthreads();
  float s = 0.0f;
  for (int i = 0; i < On; ++i) s += se[i];
  float p = e / s;
  outp[(size_t)b * On + o] = p;
  errp[(size_t)b * On + o] = p - ((y[b] == o) ? 1.0f : 0.0f);
}
__global__ void proj_kernel(const float* __restrict__ err,
                            const float* B0, const float* B1, const float* B2,
                            const float* B3, const float* B4, const float* B5,
                            float* __restrict__ proj) {
  int g = blockIdx.y;
  const float* Bm = (g == 0) ? B0 : (g == 1) ? B1 : (g == 2) ? B2
                   : (g == 3) ? B3 : (g == 4) ? B4 : B5;
  int idx = blockIdx.x * blockDim.x + threadIdx.x;
  int b = idx >> 9, hc = idx & (Hn - 1);
  float s = 0.0f;
  const float* e = err + (size_t)b * On;
  for (int o = 0; o < On; ++o) s += e[o] * Bm[(size_t)o * Hn + hc];
  proj[(size_t)g * BH + idx] = s;
}
__global__ void make_zg_kernel(float* __restrict__ outp, const float* __restrict__ proj,
                               const float* __restrict__ hp, const float* __restrict__ hs,
                               const float* __restrict__ rs, int n) {
  int i = blockIdx.x * blockDim.x + threadIdx.x;
  if (i >= n) return;
  int hc = i & (Hn - 1), b = (i >> 9) & (Bsz - 1);
  float r = rs[i];
  outp[i] = (hp[i] - hs[i]) * r * (1.0f - r) * proj[(size_t)b * Hn + hc];
}
__global__ void make_g3_kernel(float* __restrict__ outp, const float* __restrict__ proj,
                               const float* __restrict__ zs, const float* __restrict__ hs,
                               int n) {
  int i = blockIdx.x * blockDim.x + threadIdx.x;
  if (i >= n) return;
  int hc = i & (Hn - 1), b = (i >> 9) & (Bsz - 1);
  float h = hs[i];
  outp[i] = proj[(size_t)b * Hn + hc] * (1.0f - zs[i]) * (1.0f - h * h);
}
__global__ void make_g1_kernel(float* __restrict__ outp, const float* __restrict__ proj,
                               const float* __restrict__ zs, int n) {
  int i = blockIdx.x * blockDim.x + threadIdx.x;
  if (i >= n) return;
  int hc = i & (Hn - 1), b = (i >> 9) & (Bsz - 1);
  outp[i] = proj[(size_t)b * Hn + hc] * (1.0f - zs[i]);
}
__global__ void mul_hprr_kernel(float* __restrict__ P, const float* __restrict__ hp,
                                const float* __restrict__ rs, int n) {
  int i = blockIdx.x * blockDim.x + threadIdx.x;
  if (i >= n) return;
  float r = rs[i];
  P[i] *= hp[i] * r * (1.0f - r);
}
__global__ void mul_dh_kernel(float* __restrict__ P, const float* __restrict__ hs, int n) {
  int i = blockIdx.x * blockDim.x + threadIdx.x;
  if (i >= n) return;
  float h = hs[i];
  P[i] *= (1.0f - h * h);
}
__global__ void make_rh_kernel(float* __restrict__ outp, const float* __restrict__ rs,
                               const float* __restrict__ hp, int n) {
  int i = blockIdx.x * blockDim.x + threadIdx.x;
  if (i >= n) return;
  outp[i] = rs[i] * hp[i];
}
__global__ void colsum_kernel(const float* __restrict__ G, float* __restrict__ outp,
                              int rows, int cols, float scale) {
  int col = blockIdx.x;
  float s = 0.0f;
  for (int m = threadIdx.x; m < rows; m += blockDim.x) s += G[(size_t)m * cols + col];
  __shared__ float sm[256];
  sm[threadIdx.x] = s; __syncthreads();
  for (int w = 128; w > 0; w >>= 1) {
    if (threadIdx.x < w) sm[threadIdx.x] += sm[threadIdx.x + w];
    __syncthreads();
  }
  if (threadIdx.x == 0) outp[col] = sm[0] * scale;
}
__global__ void sumsq_kernel(const float* __restrict__ g, int n, float* __restrict__ part) {
  float s = 0.0f;
  for (int i = blockIdx.x * blockDim.x + threadIdx.x; i < n; i += gridDim.x * blockDim.x) {
    float v = g[i]; s += v * v;
  }
  __shared__ float sm[256];
  sm[threadIdx.x] = s; __syncthreads();
  for (int w = 128; w > 0; w >>= 1) {
    if (threadIdx.x < w) sm[threadIdx.x] += sm[threadIdx.x + w];
    __syncthreads();
  }
  if (threadIdx.x == 0) part[blockIdx.x] = sm[0];
}
__global__ void normclip_kernel(float* __restrict__ g, int n,
                                const float* __restrict__ part, int np) {
  __shared__ float norm;
  if (threadIdx.x == 0) {
    float s = 0.0f;
    for (int i = 0; i < np; ++i) s += part[i];
    norm = sqrtf(s);
  }
  __syncthreads();
  float inv = 1.0f / norm;
  for (int i = blockIdx.x * blockDim.x + threadIdx.x; i < n; i += gridDim.x * blockDim.x) {
    float v = g[i] * inv;
    g[i] = fminf(fmaxf(v, -5.0f), 5.0f);
  }
}

// ---------------- Host orchestration ----------------
extern "C" void kernel_launch(void* const* d_in, const int* in_sizes, int n_in,
                              void* d_out_v, int out_size, void* d_ws, size_t ws_size,
                              hipStream_t stream) {
  const float* x     = (const float*)d_in[0];
  const int*   y     = (const int*)d_in[1];
  const float* W1w   = (const float*)d_in[2];
  const float* V1w   = (const float*)d_in[3];
  const float* V1b   = (const float*)d_in[4];
  const float* W2w   = (const float*)d_in[5];
  const float* W2b   = (const float*)d_in[6];
  const float* V2w   = (const float*)d_in[7];
  const float* W3w   = (const float*)d_in[8];
  const float* W3b   = (const float*)d_in[9];
  const float* V3w   = (const float*)d_in[10];
  const float* Woutw = (const float*)d_in[11];
  const float* Woutb = (const float*)d_in[12];
  const float* BW1   = (const float*)d_in[13];
  const float* BV1   = (const float*)d_in[14];
  const float* BW2   = (const float*)d_in[15];
  const float* BV2   = (const float*)d_in[16];
  const float* BW3   = (const float*)d_in[17];
  const float* BV3   = (const float*)d_in[18];

  float* outb = (float*)d_out_v;
  float* ws   = (float*)d_ws;

  // workspace layout (floats)
  float* xmat = ws;                         // TB*In
  float* s0   = ws + 2097152;               // TBH (Xv1, then G temps)
  float* s1   = s0 + TBH;                   // TBH (Xv2, then P temps)
  float* s2   = s1 + TBH;                   // TBH (Xv3, then hprev)
  float* rs   = s2 + TBH;                   // TBH
  float* zs   = rs + TBH;                   // TBH
  float* hs   = zs + TBH;                   // TBH
  float* hid  = hs + TBH;                   // TBH
  float* h0   = hid + TBH;                  // BH
  float* hr   = h0 + BH;                    // BH
  float* proj = hr + BH;                    // 6*BH
  float* part = proj + 6 * BH;              // 64
  int*   bar  = (int*)(part + 64);          // 2 ints (barrier cnt, gen)

  // d_out offsets (return order of the reference tuple)
  float* o_out = outb;                      // [B,O]
  float* o_hT  = outb + 4096;               // [B,H]
  float* o_dW0 = outb + 36864;              // [H,H]
  float* o_dW1 = outb + 299008;
  float* o_dW2 = outb + 561152;
  float* o_dV0 = outb + 823296;             // [H,I]
  float* o_dV1 = outb + 888832;
  float* o_dV2 = outb + 954368;
  float* o_db0 = outb + 1019904;            // [H]
  float* o_db1 = outb + 1020416;
  float* o_db2 = outb + 1020928;
  float* o_err = outb + 1021440;            // [B,O]

  const float inv_b = 1.0f / (float)Bsz;
  const int ewg = TBH / 256;

  zero_kernel<<<(BH + 255) / 256, 256, 0, stream>>>(h0, BH);
  zero_kernel<<<1, 256, 0, stream>>>((float*)bar, 2);   // 0.0f bits == 0
  xmat_kernel<<<(TB * In + 255) / 256, 256, 0, stream>>>(x, xmat);

  // Hoisted input projections: Xv1 = x@V1^T + V1b, Xv2 = x@V2^T, Xv3 = x@V3^T
  gemm_abt4<<<dim3(TB / 16, Hn / 256), 128, 0, stream>>>(xmat, In, V1w, In, s0, Hn, V1b, In, 1.0f);
  gemm_abt4<<<dim3(TB / 16, Hn / 256), 128, 0, stream>>>(xmat, In, V2w, In, s1, Hn, nullptr, In, 1.0f);
  gemm_abt4<<<dim3(TB / 16, Hn / 256), 128, 0, stream>>>(xmat, In, V3w, In, s2, Hn, nullptr, In, 1.0f);

  // Persistent serial GRU scan: 64 blocks x 128 threads, LDS-staged gate weights
  scan_kernel<<<64, 128, 64 * LDS_STRIDE * sizeof(float), stream>>>(
      W1w, W2w, W2b, W3w, W3b, s0, s1, s2, zs, rs, hs, hid, h0, hr, bar, bar + 1);

  const float* hT = hid + (size_t)(Tn - 1) * BH;
  head_kernel<<<Bsz, On, 0, stream>>>(hT, Woutw, Woutb, y, o_out, o_err);
  copy_kernel<<<(BH + 255) / 256, 256, 0, stream>>>(hT, o_hT, BH);
  proj_kernel<<<dim3(BH / 256, 6), 256, 0, stream>>>(o_err, BW1, BV1, BW2, BV2, BW3, BV3, proj);
  hprev_kernel<<<ewg, 256, 0, stream>>>(hid, s2);   // s2 now = hprev

  // ---- z-gate grads (index 1): zg = (hprev-hs)*r*(1-r) ----
  make_zg_kernel<<<ewg, 256, 0, stream>>>(s0, proj + 3 * BH /*BV2e*/, s2, hs, rs, TBH);
  gemm_atb<<<dim3(Hn / 16, In / 64), 128, 0, stream>>>(s0, Hn, xmat, In, o_dV1, In, TB, inv_b);
  colsum_kernel<<<Hn, 256, 0, stream>>>(s0, o_db1, TB, Hn, inv_b);
  make_zg_kernel<<<ewg, 256, 0, stream>>>(s0, proj + 2 * BH /*BW2e*/, s2, hs, rs, TBH);
  gemm_atb4<<<dim3(Hn / 16, Hn / 256), 128, 0, stream>>>(s0, Hn, s2, Hn, o_dW1, Hn, TB, inv_b);

  // ---- r-gate grads (index 2) ----
  make_g3_kernel<<<ewg, 256, 0, stream>>>(s0, proj + 5 * BH /*BV3e*/, zs, hs, TBH);
  gemm_abt4<<<dim3(TB / 16, Hn / 256), 128, 0, stream>>>(s0, Hn, W1w, Hn, s1, Hn, nullptr, Hn, 1.0f);
  mul_hprr_kernel<<<ewg, 256, 0, stream>>>(s1, s2, rs, TBH);
  gemm_atb<<<dim3(Hn / 16, In / 64), 128, 0, stream>>>(s1, Hn, xmat, In, o_dV2, In, TB, inv_b);
  colsum_kernel<<<Hn, 256, 0, stream>>>(s1, o_db2, TB, Hn, inv_b);
  make_g3_kernel<<<ewg, 256, 0, stream>>>(s0, proj + 4 * BH /*BW3e*/, zs, hs, TBH);
  gemm_abt4<<<dim3(TB / 16, Hn / 256), 128, 0, stream>>>(s0, Hn, W1w, Hn, s1, Hn, nullptr, Hn, 1.0f);
  mul_hprr_kernel<<<ewg, 256, 0, stream>>>(s1, s2, rs, TBH);
  gemm_atb4<<<dim3(Hn / 16, Hn / 256), 128, 0, stream>>>(s1, Hn, s2, Hn, o_dW2, Hn, TB, inv_b);

  // ---- h-gate grads (index 0) ----
  make_g1_kernel<<<ewg, 256, 0, stream>>>(s0, proj + 1 * BH /*BV1e*/, zs, TBH);
  gemm_abt4<<<dim3(TB / 16, Hn / 256), 128, 0, stream>>>(s0, Hn, W1w, Hn, s1, Hn, nullptr, Hn, 1.0f);
  mul_dh_kernel<<<ewg, 256, 0, stream>>>(s1, hs, TBH);
  gemm_atb<<<dim3(Hn / 16, In / 64), 128, 0, stream>>>(s1, Hn, xmat, In, o_dV0, In, TB, inv_b);
  colsum_kernel<<<Hn, 256, 0, stream>>>(s1, o_db0, TB, Hn, inv_b);
  make_g1_kernel<<<ewg, 256, 0, stream>>>(s0, proj + 0 * BH /*BW1e*/, zs, TBH);
  gemm_abt4<<<dim3(TB / 16, Hn / 256), 128, 0, stream>>>(s0, Hn, W1w, Hn, s1, Hn, nullptr, Hn, 1.0f);
  mul_dh_kernel<<<ewg, 256, 0, stream>>>(s1, hs, TBH);
  make_rh_kernel<<<ewg, 256, 0, stream>>>(s0, rs, s2, TBH);   // rs*hprev
  gemm_atb4<<<dim3(Hn / 16, Hn / 256), 128, 0, stream>>>(s1, Hn, s0, Hn, o_dW0, Hn, TB, inv_b);

  // ---- deterministic norm + clip on the 9 gradient tensors ----
  float* gp[9] = {o_dW0, o_dW1, o_dW2, o_dV0, o_dV1, o_dV2, o_db0, o_db1, o_db2};
  int    gn[9] = {Hn * Hn, Hn * Hn, Hn * Hn, Hn * In, Hn * In, Hn * In, Hn, Hn, Hn};
  for (int i = 0; i < 9; ++i) {
    sumsq_kernel<<<64, 256, 0, stream>>>(gp[i], gn[i], part);
    normclip_kernel<<<(gn[i] + 255) / 256, 256, 0, stream>>>(gp[i], gn[i], part, 64);
  }
}